// IGLoss_12163347383010
// MI455X (gfx1250) — compile-verified
//
#include <hip/hip_runtime.h>

typedef __attribute__((ext_vector_type(2))) float v2f;
typedef __attribute__((ext_vector_type(4))) float v4f;
typedef __attribute__((ext_vector_type(8))) float v8f;

#define B_DIM 4096
#define D_DIM 1024
#define C_DIM 10000
#define N_TILES 625          // 10000 / 16
#define M_BLKS 64            // 4096 / 64  (each wave covers 64 rows)
#define ALPHA_F 0.1f
#define LAMBDA_F 0.01f

// ---------------------------------------------------------------------------
// Kernel 1: per-batch-row stats. One block per row (4096 blocks x 256 thr).
//   xx[b]   = sum_d feat[b][d]^2
//   part[b] = sum_d (feat[b][d] - means[labels[b]][d])^2
// Deterministic tree reduction (no float atomics).
// ---------------------------------------------------------------------------
__global__ __launch_bounds__(256) void rowstats_kernel(
    const float* __restrict__ feat, const long long* __restrict__ labels,
    const float* __restrict__ means, float* __restrict__ xx,
    float* __restrict__ part) {
  __shared__ float s1[256];
  __shared__ float s2[256];
  const int b = blockIdx.x;
  const int t = threadIdx.x;
  const float* frow = feat + (size_t)b * D_DIM;
  const float* mrow = means + (size_t)labels[b] * D_DIM;
  // D_DIM = 1024 = 256 threads * 4 elements, exactly one float4 per thread
  v4f f = *(const v4f*)(frow + t * 4);
  v4f m = *(const v4f*)(mrow + t * 4);
  float sxx = f.x * f.x + f.y * f.y + f.z * f.z + f.w * f.w;
  float dx = f.x - m.x, dy = f.y - m.y, dz = f.z - m.z, dw = f.w - m.w;
  float sdd = dx * dx + dy * dy + dz * dz + dw * dw;
  s1[t] = sxx;
  s2[t] = sdd;
  __syncthreads();
  for (int s = 128; s > 0; s >>= 1) {
    if (t < s) {
      s1[t] += s1[t + s];
      s2[t] += s2[t + s];
    }
    __syncthreads();
  }
  if (t == 0) {
    xx[b] = s1[0];
    part[b] = s2[0];
  }
}

// ---------------------------------------------------------------------------
// Kernel 2: per-class stats. One block per class (10000 blocks x 256 thr).
//   yy[c] = sum_d means[c][d]^2
// ---------------------------------------------------------------------------
__global__ __launch_bounds__(256) void colstats_kernel(
    const float* __restrict__ means, float* __restrict__ yy) {
  __shared__ float s1[256];
  const int c = blockIdx.x;
  const int t = threadIdx.x;
  v4f m = *(const v4f*)(means + (size_t)c * D_DIM + t * 4);
  s1[t] = m.x * m.x + m.y * m.y + m.z * m.z + m.w * m.w;
  __syncthreads();
  for (int s = 128; s > 0; s >>= 1) {
    if (t < s) s1[t] += s1[t + s];
    __syncthreads();
  }
  if (t == 0) yy[c] = s1[0];
}

// ---------------------------------------------------------------------------
// Kernel 3: reduce 4096 loss partials -> scalar (one block, deterministic).
//   loss = LAMBDA * 0.5 * sum(part) / B
// ---------------------------------------------------------------------------
__global__ __launch_bounds__(1024) void loss_finalize_kernel(
    const float* __restrict__ part, float* __restrict__ loss_out) {
  __shared__ float s1[1024];
  const int t = threadIdx.x;
  s1[t] = part[t] + part[t + 1024] + part[t + 2048] + part[t + 3072];
  __syncthreads();
  for (int s = 512; s > 0; s >>= 1) {
    if (t < s) s1[t] += s1[t + s];
    __syncthreads();
  }
  if (t == 0) loss_out[0] = LAMBDA_F * 0.5f * s1[0] / (float)B_DIM;
}

// ---------------------------------------------------------------------------
// Kernel 4: WMMA f32 GEMM + fused epilogue, register-blocked.
// One wave32 computes a 64(M) x 16(N) output slab = four 16x16 WMMA tiles
// sharing one B fragment per K-step:
//   per K-step of 4: 1x B load + 4x A loads + 4x independent v_wmma
//   -> 1.25 b64 loads per WMMA (vs 2.0 unblocked), independent accumulators.
//
// A layout (32-bit 16x4): lanes 0-15 M=lane, VGPR{0,1} = K{0,1};
//                         lanes 16-31 M=lane-16, VGPR{0,1} = K{2,3}.
// -> each lane loads b64 at row feat[m0 + (lane&15)], col k + 2*(lane>>4).
// B (4x16, means.T) mirrors this with N=lane&15 reading means rows.
// C/D layout: VGPR j, lanes 0-15 -> M=j, N=lane; lanes 16-31 -> M=j+8.
//
// Epilogue: out[m][n] = (XY - 0.5*XX[m] - 0.5*YY[n]) * var[n]^2
//                       * (labels[m]==n ? 1+ALPHA : 1)
// ---------------------------------------------------------------------------
__global__ __launch_bounds__(256) void gemm_igloss_kernel(
    const float* __restrict__ feat, const float* __restrict__ means,
    const long long* __restrict__ labels, const float* __restrict__ variance,
    const float* __restrict__ xx, const float* __restrict__ yy,
    float* __restrict__ out) {
  const int wave = (int)blockIdx.x * 8 + ((int)threadIdx.x >> 5);
  const int lane = (int)threadIdx.x & 31;
  const int r = lane & 15;
  const int hi = lane >> 4;

  const int ntile = wave % N_TILES;   // consecutive waves share A rows
  const int mblk = wave / N_TILES;
  const int m0 = mblk * 64;
  const int n0 = ntile * 16;

  const float* arow = feat + (size_t)(m0 + r) * D_DIM + hi * 2;
  const float* brow = means + (size_t)(n0 + r) * D_DIM + hi * 2;

  v8f acc[4] = {v8f{}, v8f{}, v8f{}, v8f{}};
#pragma unroll 2
  for (int k = 0; k < D_DIM; k += 4) {
    v2f b = *(const v2f*)(brow + k);
    v2f a0 = *(const v2f*)(arow + k);
    v2f a1 = *(const v2f*)(arow + 16 * D_DIM + k);
    v2f a2 = *(const v2f*)(arow + 32 * D_DIM + k);
    v2f a3 = *(const v2f*)(arow + 48 * D_DIM + k);
    acc[0] = __builtin_amdgcn_wmma_f32_16x16x4_f32(false, a0, false, b,
                                                   (short)0, acc[0], false, false);
    acc[1] = __builtin_amdgcn_wmma_f32_16x16x4_f32(false, a1, false, b,
                                                   (short)0, acc[1], false, false);
    acc[2] = __builtin_amdgcn_wmma_f32_16x16x4_f32(false, a2, false, b,
                                                   (short)0, acc[2], false, false);
    acc[3] = __builtin_amdgcn_wmma_f32_16x16x4_f32(false, a3, false, b,
                                                   (short)0, acc[3], false, false);
  }

  const int n = n0 + r;
  float vv = variance[n];
  vv *= vv;
  const float yyh = 0.5f * yy[n];

#pragma unroll
  for (int i = 0; i < 4; ++i) {
    const int mbase = m0 + i * 16 + (hi ? 8 : 0);
#pragma unroll
    for (int j = 0; j < 8; ++j) {
      const int m = mbase + j;
      float val = (acc[i][j] - 0.5f * xx[m] - yyh) * vv;
      if (labels[m] == (long long)n) val *= (1.0f + ALPHA_F);
      out[(size_t)m * C_DIM + n] = val;
    }
  }
}

// ---------------------------------------------------------------------------
extern "C" void kernel_launch(void* const* d_in, const int* in_sizes, int n_in,
                              void* d_out, int out_size, void* d_ws,
                              size_t ws_size, hipStream_t stream) {
  (void)in_sizes; (void)n_in; (void)out_size; (void)ws_size;

  const float* feat = (const float*)d_in[0];
  const long long* labels = (const long long*)d_in[1];
  const float* means = (const float*)d_in[2];
  const float* variance = (const float*)d_in[3];

  float* out = (float*)d_out;
  float* logits = out;                                   // B*C floats
  float* loss = out + (size_t)B_DIM * C_DIM;             // 1 float
  float* means_out = loss + 1;                           // C*D floats

  float* ws = (float*)d_ws;
  float* xx = ws;                  // [4096]
  float* yy = ws + B_DIM;          // [10000]
  float* part = yy + C_DIM;        // [4096]

  rowstats_kernel<<<B_DIM, 256, 0, stream>>>(feat, labels, means, xx, part);
  colstats_kernel<<<C_DIM, 256, 0, stream>>>(means, yy);
  loss_finalize_kernel<<<1, 1024, 0, stream>>>(part, loss);

  const int total_waves = M_BLKS * N_TILES;  // 40000
  gemm_igloss_kernel<<<total_waves / 8, 256, 0, stream>>>(
      feat, means, labels, variance, xx, yy, logits);

  hipMemcpyAsync(means_out, means, (size_t)C_DIM * D_DIM * sizeof(float),
                 hipMemcpyDeviceToDevice, stream);
}